// EntropyBottleneck_62758062129742
// MI455X (gfx1250) — compile-verified
//
#include <hip/hip_runtime.h>
#include <math.h>

// Problem constants (match reference)
#define BB   8
#define CC   192
#define HH   128
#define WWD  128
#define HWP  (HH * WWD)                 // 16384
#define TOTAL ((size_t)BB * CC * HWP)   // 25165824 elements per output tensor

typedef float v2f __attribute__((ext_vector_type(2)));
typedef float v8f __attribute__((ext_vector_type(8)));

#define LOG2E 1.4426950408889634f

// ---- fast native transcendentals (gfx1250 TRANS unit) ----
__device__ __forceinline__ float fast_exp2(float x) {
#if __has_builtin(__builtin_amdgcn_exp2f)
  return __builtin_amdgcn_exp2f(x);
#else
  return exp2f(x);
#endif
}
__device__ __forceinline__ float fast_rcp(float x) {
#if __has_builtin(__builtin_amdgcn_rcpf)
  return __builtin_amdgcn_rcpf(x);
#else
  return 1.0f / x;
#endif
}
__device__ __forceinline__ float fast_tanh(float x) {
#if __has_builtin(__builtin_amdgcn_tanhf)
  return __builtin_amdgcn_tanhf(x);  // v_tanh_f32
#else
  // tanh(x) = 1 - 2/(exp(2x)+1); exp via native exp2. Saturates correctly:
  // x >> 0 -> exp2 -> +inf -> rcp -> 0 -> 1 ; x << 0 -> exp2 -> 0 -> -1.
  float e = fast_exp2(x * (2.0f * LOG2E));
  return fmaf(-2.0f, fast_rcp(e + 1.0f), 1.0f);
#endif
}
__device__ __forceinline__ float fast_sigmoid(float x) {
  // 1/(1+exp(-x)) with native exp2 + rcp
  return fast_rcp(1.0f + fast_exp2(-x * LOG2E));
}

// ---- float <-> orderable uint (for uint atomic min/max of floats) ----
__device__ __forceinline__ unsigned orderKey(float f) {
  unsigned u = __float_as_uint(f);
  return (u & 0x80000000u) ? ~u : (u | 0x80000000u);
}
__device__ __forceinline__ float orderDecode(unsigned k) {
  return (k & 0x80000000u) ? __uint_as_float(k & 0x7fffffffu)
                           : __uint_as_float(~k);
}
__device__ __forceinline__ float softplusf(float v) {
  // numerically stable softplus, matches jax.nn.softplus (runs once; libm ok)
  return fmaxf(v, 0.0f) + log1pf(expf(-fabsf(v)));
}

// ---------------------------------------------------------------------------
// Kernel 1: preprocess per-channel parameters + init min/max atomic slots.
// ws layout: [0..15] uint32 (min keys b=0..7, max keys b=0..7),
//            then 192 channels * 64 floats of packed params:
//  [0..2]  softplus(m0)          [3..11]  softplus(m1) row-major
//  [12..20] softplus(m2)         [21..29] softplus(m3)
//  [30..32] softplus(m4)         [33..35] b0  [36..38] b1
//  [39..41] b2  [42..44] b3      [45] b4
//  [46..48] tanh(f0) [49..51] tanh(f1) [52..54] tanh(f2) [55..57] tanh(f3)
// ---------------------------------------------------------------------------
__global__ void eb_prep(const float* __restrict__ m0, const float* __restrict__ b0,
                        const float* __restrict__ m1, const float* __restrict__ b1,
                        const float* __restrict__ m2, const float* __restrict__ b2,
                        const float* __restrict__ m3, const float* __restrict__ b3,
                        const float* __restrict__ m4, const float* __restrict__ b4,
                        const float* __restrict__ f0, const float* __restrict__ f1,
                        const float* __restrict__ f2, const float* __restrict__ f3,
                        unsigned* __restrict__ mm, float* __restrict__ P) {
  int t = threadIdx.x;
  if (t < CC) {
    float* p = P + t * 64;
    for (int j = 0; j < 3; ++j) p[j] = softplusf(m0[t * 3 + j]);
    const float* ms[3] = {m1, m2, m3};
    for (int l = 0; l < 3; ++l)
      for (int j = 0; j < 9; ++j) p[3 + l * 9 + j] = softplusf(ms[l][t * 9 + j]);
    for (int k = 0; k < 3; ++k) p[30 + k] = softplusf(m4[t * 3 + k]);
    const float* bs[4] = {b0, b1, b2, b3};
    for (int l = 0; l < 4; ++l)
      for (int j = 0; j < 3; ++j) p[33 + l * 3 + j] = bs[l][t * 3 + j];
    p[45] = b4[t];
    const float* fs[4] = {f0, f1, f2, f3};
    for (int l = 0; l < 4; ++l)
      for (int j = 0; j < 3; ++j) p[46 + l * 3 + j] = tanhf(fs[l][t * 3 + j]);
    for (int j = 58; j < 64; ++j) p[j] = 0.0f;
  } else if (t < CC + 16) {
    int i = t - CC;
    mm[i] = (i < 8) ? 0xFFFFFFFFu : 0u;  // min slots := +inf key, max := -inf key
  }
}

// ---------------------------------------------------------------------------
// Kernel 2: per-batch min/max reduction (wave shuffle + LDS + one atomic/block)
// ---------------------------------------------------------------------------
__global__ __launch_bounds__(256) void eb_minmax(const float* __restrict__ x,
                                                 unsigned* __restrict__ mm) {
  const int per = CC * HWP;
  const int b = blockIdx.y;
  const float* xb = x + (size_t)b * per;
  float mn = INFINITY, mx = -INFINITY;
  for (int i = blockIdx.x * blockDim.x + threadIdx.x; i < per;
       i += gridDim.x * blockDim.x) {
    float v = xb[i];
    mn = fminf(mn, v);
    mx = fmaxf(mx, v);
  }
  for (int o = 16; o; o >>= 1) {
    mn = fminf(mn, __shfl_xor(mn, o, 32));
    mx = fmaxf(mx, __shfl_xor(mx, o, 32));
  }
  __shared__ float smn[8], smx[8];
  const int w = threadIdx.x >> 5, lane = threadIdx.x & 31;
  if (lane == 0) { smn[w] = mn; smx[w] = mx; }
  __syncthreads();
  if (threadIdx.x == 0) {
    for (int i = 1; i < 8; ++i) { mn = fminf(mn, smn[i]); mx = fmaxf(mx, smx[i]); }
    atomicMin(&mm[b], orderKey(mn));
    atomicMax(&mm[8 + b], orderKey(mx));
  }
}

// ---------------------------------------------------------------------------
// Kernel 3: main. Per wave-iteration: 8 elements x {lower,upper} = 16 WMMA
// columns. A holds softplus(M) duplicated in rows 0-2 and 8-10 so the D
// matrix's upper half mirrors its lower half -> lanes 16-31 always mirror
// lanes 0-15 and the next B operand is a lane-local select (no shuffles).
// Gates use native v_tanh_f32 (1 TRANS op) so the inner loop is a tight
// WMMA + TRANS pipeline with no EXEC divergence.
// ---------------------------------------------------------------------------
__global__ __launch_bounds__(256) void eb_main(const float* __restrict__ x,
                                               const unsigned* __restrict__ mm,
                                               const float* __restrict__ P,
                                               float* __restrict__ out) {
  const int c = blockIdx.y, b = blockIdx.z;
  const int tid = threadIdx.x;
  const int lane = tid & 31, wave = tid >> 5;
  const float* p = P + c * 64;

  const float mn = orderDecode(mm[b]);
  const float mx = orderDecode(mm[8 + b]);
  const float qs = 65535.0f / (mx - mn + 1e-12f);

  // A operands for the three 3x3 WMMA layers (16x4 f32 layout:
  // lanes 0-15 -> M=0..15 with {K0,K1}, lanes 16-31 -> M=0..15 with {K2,K3})
  const int m = lane & 15, j = m & 7;
  const bool hi = lane >= 16;
  const bool rowok = (j < 3);  // weight rows live at M = 0..2 and 8..10
  v2f A1, A2, A3;
  {
    const float* s1 = p + 3;
    const float* s2 = p + 12;
    const float* s3 = p + 21;
    const int k0 = hi ? 2 : 0;
    A1[0] = rowok ? s1[j * 3 + k0] : 0.0f;
    A1[1] = (rowok && !hi) ? s1[j * 3 + 1] : 0.0f;  // K=3 column is zero
    A2[0] = rowok ? s2[j * 3 + k0] : 0.0f;
    A2[1] = (rowok && !hi) ? s2[j * 3 + 1] : 0.0f;
    A3[0] = rowok ? s3[j * 3 + k0] : 0.0f;
    A3[1] = (rowok && !hi) ? s3[j * 3 + 1] : 0.0f;
  }
  // Bias baked into the C accumulator: C/D VGPR r holds rows r and r+8,
  // both of which carry bias[r] for r<3, zero elsewhere (uniform per VGPR).
  v8f Cb1 = {0, 0, 0, 0, 0, 0, 0, 0};
  v8f Cb2 = Cb1, Cb3 = Cb1;
  Cb1[0] = p[36]; Cb1[1] = p[37]; Cb1[2] = p[38];
  Cb2[0] = p[39]; Cb2[1] = p[40]; Cb2[2] = p[41];
  Cb3[0] = p[42]; Cb3[1] = p[43]; Cb3[2] = p[44];

  const float s00 = p[0], s01 = p[1], s02 = p[2];
  const float bb00 = p[33], bb01 = p[34], bb02 = p[35];
  const float s40 = p[30], s41 = p[31], s42 = p[32], bb4 = p[45];
  const float t00 = p[46], t01 = p[47], t02 = p[48];
  const float t10 = p[49], t11 = p[50], t12 = p[51];
  const float t20 = p[52], t21 = p[53], t22 = p[54];
  const float t30 = p[55], t31 = p[56], t32 = p[57];

  const size_t plane = ((size_t)b * CC + c) * HWP;
  // columns 0-7: lower pass (x-0.5); columns 8-15: upper pass (x+0.5);
  // lanes 16-31 mirror columns 0-15.
  const float half = ((lane & 15) < 8) ? -0.5f : 0.5f;
  const int eBase = blockIdx.x * 512 + wave * 64;

#pragma unroll 2
  for (int t = 0; t < 8; ++t) {
    const int e = eBase + t * 8 + (lane & 7);
    const float v = x[plane + e];
    const float xd = rintf((v - mn) * qs) * (1.0f / 65535.0f);
    const float xin = xd + half;

    // layer 0: (3x1) @ (1xN) + b0, then tanh gate
    float g0 = fmaf(s00, xin, bb00);
    float g1 = fmaf(s01, xin, bb01);
    float g2 = fmaf(s02, xin, bb02);
    g0 = fmaf(t00, fast_tanh(g0), g0);
    g1 = fmaf(t01, fast_tanh(g1), g1);
    g2 = fmaf(t02, fast_tanh(g2), g2);

    v2f Bv; v8f d;
    // layer 1 (WMMA): B 4x16 layout -> VGPR0: rows {0,2}, VGPR1: rows {1,3}
    Bv[0] = hi ? g2 : g0; Bv[1] = hi ? 0.0f : g1;
    d = __builtin_amdgcn_wmma_f32_16x16x4_f32(false, A1, false, Bv,
                                              (short)0, Cb1, false, false);
    g0 = d[0]; g1 = d[1]; g2 = d[2];
    g0 = fmaf(t10, fast_tanh(g0), g0);
    g1 = fmaf(t11, fast_tanh(g1), g1);
    g2 = fmaf(t12, fast_tanh(g2), g2);

    // layer 2 (WMMA)
    Bv[0] = hi ? g2 : g0; Bv[1] = hi ? 0.0f : g1;
    d = __builtin_amdgcn_wmma_f32_16x16x4_f32(false, A2, false, Bv,
                                              (short)0, Cb2, false, false);
    g0 = d[0]; g1 = d[1]; g2 = d[2];
    g0 = fmaf(t20, fast_tanh(g0), g0);
    g1 = fmaf(t21, fast_tanh(g1), g1);
    g2 = fmaf(t22, fast_tanh(g2), g2);

    // layer 3 (WMMA)
    Bv[0] = hi ? g2 : g0; Bv[1] = hi ? 0.0f : g1;
    d = __builtin_amdgcn_wmma_f32_16x16x4_f32(false, A3, false, Bv,
                                              (short)0, Cb3, false, false);
    g0 = d[0]; g1 = d[1]; g2 = d[2];
    g0 = fmaf(t30, fast_tanh(g0), g0);
    g1 = fmaf(t31, fast_tanh(g1), g1);
    g2 = fmaf(t32, fast_tanh(g2), g2);

    // layer 4: (1x3) dot, then sigmoid
    const float logit = fmaf(s40, g0, fmaf(s41, g1, fmaf(s42, g2, bb4)));
    const float sig = fast_sigmoid(logit);

    // likelihood = sigmoid(upper) - sigmoid(lower): partner lives at lane|8
    const float sigU = __shfl(sig, lane | 8, 32);
    const float sigL = __shfl(sig, lane & ~8, 32);
    const float lik = fmaxf(sigU - sigL, 1e-9f);

    if (lane < 8) {  // one writer per element (divergence only after WMMAs)
      const size_t idx = plane + e;
      out[idx] = xd;
      out[TOTAL + idx] = lik;
    }
  }
}

// ---------------------------------------------------------------------------
extern "C" void kernel_launch(void* const* d_in, const int* in_sizes, int n_in,
                              void* d_out, int out_size, void* d_ws, size_t ws_size,
                              hipStream_t stream) {
  (void)in_sizes; (void)n_in; (void)out_size; (void)ws_size;
  // dict order: x, m0,b0, m1,b1, m2,b2, m3,b3, m4,b4, f0,f1,f2,f3
  const float* x  = (const float*)d_in[0];
  const float* m0 = (const float*)d_in[1];
  const float* b0 = (const float*)d_in[2];
  const float* m1 = (const float*)d_in[3];
  const float* b1 = (const float*)d_in[4];
  const float* m2 = (const float*)d_in[5];
  const float* b2 = (const float*)d_in[6];
  const float* m3 = (const float*)d_in[7];
  const float* b3 = (const float*)d_in[8];
  const float* m4 = (const float*)d_in[9];
  const float* b4 = (const float*)d_in[10];
  const float* f0 = (const float*)d_in[11];
  const float* f1 = (const float*)d_in[12];
  const float* f2 = (const float*)d_in[13];
  const float* f3 = (const float*)d_in[14];

  unsigned* mm = (unsigned*)d_ws;          // 16 uint slots
  float* P = (float*)d_ws + 16;            // 192 * 64 floats of packed params

  hipLaunchKernelGGL(eb_prep, dim3(1), dim3(256), 0, stream,
                     m0, b0, m1, b1, m2, b2, m3, b3, m4, b4, f0, f1, f2, f3,
                     mm, P);
  hipLaunchKernelGGL(eb_minmax, dim3(256, 8), dim3(256), 0, stream, x, mm);
  hipLaunchKernelGGL(eb_main, dim3(32, 192, 8), dim3(256), 0, stream,
                     x, mm, P, (float*)d_out);
}